// ResidualVectorQuantizer_84653805404466
// MI455X (gfx1250) — compile-verified
//
#include <hip/hip_runtime.h>

typedef float v2f __attribute__((ext_vector_type(2)));
typedef float v8f __attribute__((ext_vector_type(8)));

#define LEVELS 4
#define KCB    1024
#define DIM    64
#define NROWS  (16 * 8192)          // 131072 vectors
#define ROWS_PER_BLOCK 128
#define BLOCK  128                  // 4 wave32; each wave owns 2 x 16-row tiles
#define NWAVES 4
#define RPAD 65                     // padded LDS row stride (floats)
#define CPAD 65

// ---------------------------------------------------------------------------
// Kernel 0: half-norms 0.5*||c||^2 per code, zero counts + sumsq accumulators
// ---------------------------------------------------------------------------
__global__ __launch_bounds__(256) void rvq_prep(const float* __restrict__ cbs,
                                                float* __restrict__ hn,
                                                float* __restrict__ counts,
                                                float* __restrict__ sumsq) {
  int tid = blockIdx.x * 256 + threadIdx.x;   // 0..4095 == level*1024 + code
  const float* c = cbs + (long)tid * DIM;
  float s = 0.f;
#pragma unroll
  for (int d = 0; d < DIM; ++d) { float v = c[d]; s += v * v; }
  hn[tid]     = 0.5f * s;
  counts[tid] = 0.f;
  if (tid < LEVELS) sumsq[tid] = 0.f;
}

// ---------------------------------------------------------------------------
// Kernel 1: fused 4-level RVQ. Block = 128 threads (4 wave32), 128 rows.
// Each wave owns two 16-row A tiles (B fragment reuse: 1 ds_load -> 2 WMMAs).
// Codebook tiles double-buffered through registers + LDS (1 barrier / tile).
// ---------------------------------------------------------------------------
__global__ __launch_bounds__(BLOCK) void rvq_main(const float* __restrict__ x,
                                                  const float* __restrict__ cbs,
                                                  const float* __restrict__ hn,
                                                  float* __restrict__ counts,
                                                  float* __restrict__ sumsq,
                                                  float* __restrict__ out_q,
                                                  float* __restrict__ out_codes) {
  __shared__ float resid[ROWS_PER_BLOCK * RPAD];  // 128 x 64 residual (padded)
  __shared__ float cbt[2 * 16 * CPAD];            // double-buffered 16-code tile
  __shared__ int   widx[ROWS_PER_BLOCK];
  __shared__ float redbuf[NWAVES];

  const int t    = threadIdx.x;
  const int lane = t & 31;
  const int wave = t >> 5;
  const int half = lane >> 4;       // 0: lanes 0-15, 1: lanes 16-31
  const int l15  = lane & 15;
  const long rowbase = (long)blockIdx.x * ROWS_PER_BLOCK;
  const long ebase   = rowbase * DIM;

  // Stage coordinates for cooperative codebook-tile copy (8 floats/thread).
  const int srow = t >> 3;            // 0..15
  const int scol = (t & 7) * 8;       // 0..56

  // Load x tile into LDS residual (coalesced).
  for (int i = 0; i < 64; ++i) {
    int e = t + BLOCK * i;
    int r = e >> 6, c = e & 63;
    resid[r * RPAD + c] = x[ebase + e];
  }

  for (int lvl = 0; lvl < LEVELS; ++lvl) {
    const float* cb  = cbs + (long)lvl * KCB * DIM;
    const float* hnl = hn + lvl * KCB;
    __syncthreads();

    // A fragments for two 16-row tiles (ISA 7.12.2 32-bit A layout:
    // lanes 0-15 hold depth {4k,4k+1}, lanes 16-31 hold {4k+2,4k+3}).
    v2f a0[16], a1[16];
    {
      int r0 = wave * 32 + l15;
      int r1 = r0 + 16;
#pragma unroll
      for (int kk = 0; kk < 16; ++kk) {
        int d = 4 * kk + 2 * half;
        a0[kk].x = resid[r0 * RPAD + d];
        a0[kk].y = resid[r0 * RPAD + d + 1];
        a1[kk].x = resid[r1 * RPAD + d];
        a1[kk].y = resid[r1 * RPAD + d + 1];
      }
    }

    float bv0[8], bv1[8];
    int   bc0[8], bc1[8];
#pragma unroll
    for (int v = 0; v < 8; ++v) {
      bv0[v] = -3.4e38f; bc0[v] = 0;
      bv1[v] = -3.4e38f; bc1[v] = 0;
    }

    // Prologue: stage tile 0 into buffer 0, prefetch tile 1 into registers.
    float4 sA = *reinterpret_cast<const float4*>(cb + (long)srow * DIM + scol);
    float4 sB = *reinterpret_cast<const float4*>(cb + (long)srow * DIM + scol + 4);
    {
      float* dst = cbt + srow * CPAD + scol;
      dst[0] = sA.x; dst[1] = sA.y; dst[2] = sA.z; dst[3] = sA.w;
      dst[4] = sB.x; dst[5] = sB.y; dst[6] = sB.z; dst[7] = sB.w;
    }
    sA = *reinterpret_cast<const float4*>(cb + (long)(16 + srow) * DIM + scol);
    sB = *reinterpret_cast<const float4*>(cb + (long)(16 + srow) * DIM + scol + 4);

    for (int ct = 0; ct < 64; ++ct) {
      const int cur = ct & 1;
      __syncthreads();   // staged buf[cur] visible; buf[cur^1] free to overwrite
      if (ct + 1 < 64) {
        float* dst = cbt + (cur ^ 1) * 16 * CPAD + srow * CPAD + scol;
        dst[0] = sA.x; dst[1] = sA.y; dst[2] = sA.z; dst[3] = sA.w;
        dst[4] = sB.x; dst[5] = sB.y; dst[6] = sB.z; dst[7] = sB.w;
        if (ct + 2 < 64) {
          const float* src = cb + (long)((ct + 2) * 16 + srow) * DIM + scol;
          sA = *reinterpret_cast<const float4*>(src);
          sB = *reinterpret_cast<const float4*>(src + 4);
          __builtin_prefetch(cb + (long)((ct + 3) & 63) * 16 * DIM, 0, 1);
        }
      }

      const float* cbuf = cbt + cur * 16 * CPAD;
      const int col = ct * 16 + l15;
      const float h = hnl[col];

      // acc initialized to -0.5*||c||^2 -> WMMA produces the score directly.
      v8f acc0, acc1;
#pragma unroll
      for (int v = 0; v < 8; ++v) { acc0[v] = -h; acc1[v] = -h; }

#pragma unroll
      for (int kk = 0; kk < 16; ++kk) {
        int d = 4 * kk + 2 * half;
        v2f b;
        b.x = cbuf[l15 * CPAD + d];
        b.y = cbuf[l15 * CPAD + d + 1];
        acc0 = __builtin_amdgcn_wmma_f32_16x16x4_f32(
            false, a0[kk], false, b, (short)0, acc0, false, false);
        acc1 = __builtin_amdgcn_wmma_f32_16x16x4_f32(
            false, a1[kk], false, b, (short)0, acc1, false, false);
      }

#pragma unroll
      for (int v = 0; v < 8; ++v) {
        float s0 = acc0[v];
        float s1 = acc1[v];
        if (s0 > bv0[v]) { bv0[v] = s0; bc0[v] = col; }
        if (s1 > bv1[v]) { bv1[v] = s1; bc1[v] = col; }
      }
    }

    // Cross-lane argmax within each 16-lane half (ties -> smaller index).
#pragma unroll
    for (int v = 0; v < 8; ++v) {
#pragma unroll
      for (int off = 1; off < 16; off <<= 1) {
        float ov0 = __shfl_xor(bv0[v], off, 32);
        int   oi0 = __shfl_xor(bc0[v], off, 32);
        if (ov0 > bv0[v] || (ov0 == bv0[v] && oi0 < bc0[v])) { bv0[v] = ov0; bc0[v] = oi0; }
        float ov1 = __shfl_xor(bv1[v], off, 32);
        int   oi1 = __shfl_xor(bc1[v], off, 32);
        if (ov1 > bv1[v] || (ov1 == bv1[v] && oi1 < bc1[v])) { bv1[v] = ov1; bc1[v] = oi1; }
      }
    }

    if (l15 == 0) {                          // lanes 0 and 16 of each wave
#pragma unroll
      for (int v = 0; v < 8; ++v) {
        int r0 = wave * 32 + v + 8 * half;       // tile 0 row
        int r1 = r0 + 16;                        // tile 1 row
        widx[r0] = bc0[v];
        widx[r1] = bc1[v];
        out_codes[(long)lvl * NROWS + rowbase + r0] = (float)bc0[v];
        out_codes[(long)lvl * NROWS + rowbase + r1] = (float)bc1[v];
        atomicAdd(&counts[lvl * KCB + bc0[v]], 1.0f);
        atomicAdd(&counts[lvl * KCB + bc1[v]], 1.0f);
      }
    }
    __syncthreads();

    // residual -= cb[idx]; accumulate sum of squares of new residual
    // (rec_loss + com_loss = (1+beta) * mean(residual_post^2)).
    float ls = 0.f;
    for (int i = 0; i < 64; ++i) {
      int e = t + BLOCK * i;
      int r = e >> 6, c = e & 63;
      float q  = cb[(long)widx[r] * DIM + c];
      float nr = resid[r * RPAD + c] - q;
      resid[r * RPAD + c] = nr;
      ls += nr * nr;
    }
#pragma unroll
    for (int off = 1; off < 32; off <<= 1) ls += __shfl_xor(ls, off, 32);
    if (lane == 0) redbuf[wave] = ls;
    __syncthreads();
    if (t == 0) {
      float bs = 0.f;
#pragma unroll
      for (int w = 0; w < NWAVES; ++w) bs += redbuf[w];
      atomicAdd(&sumsq[lvl], bs);
    }
  }

  __syncthreads();
  // quantized_out = x - residual_final
  for (int i = 0; i < 64; ++i) {
    int e = t + BLOCK * i;
    int r = e >> 6, c = e & 63;
    out_q[ebase + e] = x[ebase + e] - resid[r * RPAD + c];
  }
}

// ---------------------------------------------------------------------------
// Kernel 2: finalize total loss = sum_l [(1+beta)*sumsq_l/(N*D) + 0.1*H_l]
// ---------------------------------------------------------------------------
__global__ __launch_bounds__(256) void rvq_fin(const float* __restrict__ counts,
                                               const float* __restrict__ sumsq,
                                               float* __restrict__ out_loss) {
  __shared__ float red[256];
  int t = threadIdx.x;
  float ent = 0.f;
  for (int l = 0; l < LEVELS; ++l) {
    for (int k = t; k < KCB; k += 256) {
      float p = counts[l * KCB + k] * (1.0f / (float)NROWS);
      ent -= p * logf(p + 1e-10f);
    }
  }
  red[t] = ent;
  __syncthreads();
  for (int s = 128; s > 0; s >>= 1) {
    if (t < s) red[t] += red[t + s];
    __syncthreads();
  }
  if (t == 0) {
    float tot = 0.f;
    for (int l = 0; l < LEVELS; ++l)
      tot += 1.25f * sumsq[l] * (1.0f / (float)((long)NROWS * DIM));
    tot += 0.1f * red[0];
    out_loss[0] = tot;
  }
}

// ---------------------------------------------------------------------------
extern "C" void kernel_launch(void* const* d_in, const int* in_sizes, int n_in,
                              void* d_out, int out_size, void* d_ws, size_t ws_size,
                              hipStream_t stream) {
  const float* x   = (const float*)d_in[0];    // [16,8192,64]
  const float* cbs = (const float*)d_in[1];    // [4,1024,64]

  float* ws     = (float*)d_ws;
  float* hn     = ws;                          // [4*1024]
  float* counts = ws + LEVELS * KCB;           // [4*1024]
  float* sumsq  = ws + 2 * LEVELS * KCB;       // [4]

  float* out       = (float*)d_out;
  float* out_q     = out;                      // [16*8192*64]
  float* out_codes = out + (long)NROWS * DIM;  // [4*131072]
  float* out_loss  = out + (out_size - 1);     // [1]

  rvq_prep<<<(LEVELS * KCB) / 256, 256, 0, stream>>>(cbs, hn, counts, sumsq);
  rvq_main<<<NROWS / ROWS_PER_BLOCK, BLOCK, 0, stream>>>(x, cbs, hn, counts, sumsq,
                                                         out_q, out_codes);
  rvq_fin<<<1, 256, 0, stream>>>(counts, sumsq, out_loss);
}